// PHD_conv2d_83511344103504
// MI455X (gfx1250) — compile-verified
//
#include <hip/hip_runtime.h>

typedef __attribute__((ext_vector_type(2))) float v2f;
typedef __attribute__((ext_vector_type(8))) float v8f;

#define BATCH 4
#define CIN   32
#define COUT  32
#define NLEN  81920
#define KW    10
#define CK    (CIN * KW)            // 320 reduction length
#define WPB   8                     // waves per block (main kernel)
#define COLS_PER_BLOCK (WPB * 16)   // 128 output columns per block
#define NBLK (NLEN / COLS_PER_BLOCK) // 640 blocks (batch handled inside wave)
#define TT_N  64                    // transpose tile width in N

// ---------------------------------------------------------------------------
// Kernel 1: transpose input (B, C, N) -> (B, N, C) so one gathered column
// (all 32 channels of input[:, :, idx]) occupies exactly one 128B cacheline.
// ---------------------------------------------------------------------------
__global__ __launch_bounds__(256)
void PHD_transpose_input(const float* __restrict__ in, float* __restrict__ out)
{
    __shared__ float tile[CIN][TT_N + 1];   // +1 pad: conflict-free column reads
    const int b  = blockIdx.x / (NLEN / TT_N);
    const int n0 = (blockIdx.x % (NLEN / TT_N)) * TT_N;
    const int t  = threadIdx.x;

    const float* src = in + b * (CIN * NLEN) + n0;
#pragma unroll
    for (int i = 0; i < (CIN * TT_N) / 256; ++i) {
        int e  = t + i * 256;
        int c  = e >> 6;        // /TT_N
        int nn = e & (TT_N - 1);
        tile[c][nn] = src[c * NLEN + nn];   // coalesced along n
    }
    __syncthreads();

    float* dst = out + (b * NLEN + n0) * CIN;
#pragma unroll
    for (int i = 0; i < (CIN * TT_N) / 256; ++i) {
        int e  = t + i * 256;
        int nn = e >> 5;        // /CIN
        int c  = e & (CIN - 1);
        dst[nn * CIN + c] = tile[c][nn];    // coalesced along c
    }
}

// ---------------------------------------------------------------------------
// Kernel 2: transpose weights (o, c, k) -> wt[k][m][c] so the A-operand
// lane pair (channels 2*half, 2*half+1) is contiguous -> global_load_b64.
// ---------------------------------------------------------------------------
__global__ __launch_bounds__(256)
void PHD_transpose_weight(const float* __restrict__ w, float* __restrict__ wt)
{
    int t = blockIdx.x * 256 + threadIdx.x;     // total KW*COUT*CIN = 10240
    if (t < KW * COUT * CIN) {
        int k = t / (COUT * CIN);
        int m = (t / CIN) % COUT;
        int c = t % CIN;
        wt[t] = w[m * CK + c * KW + k];
    }
}

// ---------------------------------------------------------------------------
// Kernel 3: WMMA GEMM, batch-combined. One wave computes a 32x16 output tile
// for ALL 4 batches (8 accumulators), so the batch-invariant table row and
// weight tiles are loaded once and amortized over 8 WMMAs per step.
// Reduction ordered kk = k*32 + c: one table row serves 32 kk values.
// ---------------------------------------------------------------------------
__global__ __launch_bounds__(32 * WPB)
void PHD_conv_gather_wmma_f32(const float* __restrict__ in_t,  // [b][n][c]
                              const float* __restrict__ wt,    // [k][m][c]
                              const float* __restrict__ bias,
                              const int*   __restrict__ table, // [k][n]
                              float* __restrict__ out)         // [b][o][n]
{
    const int lane = threadIdx.x & 31;
    const int wave = threadIdx.x >> 5;
    const int lcol = lane & 15;      // tile column (B) / M row (A)
    const int half = lane >> 4;      // selects K-rows {0,1} vs {2,3} of a 4-step
    const int n0   = blockIdx.x * COLS_PER_BLOCK + wave * 16;

    const int coff = 2 * half;       // channel offset of this lane's pair

    v8f acc[BATCH][2];
#pragma unroll
    for (int bb = 0; bb < BATCH; ++bb) {
        acc[bb][0] = (v8f){};
        acc[bb][1] = (v8f){};
    }

    for (int k = 0; k < KW; ++k) {
        // one coalesced, batch-invariant table load per k
        const int idx  = table[k * NLEN + n0 + lcol];
        const int boff = idx * CIN + coff;                 // shared VGPR offset
        const float* wklo = wt + k * (COUT * CIN) + lcol * CIN + coff;
#pragma unroll
        for (int c0 = 0; c0 < CIN; c0 += 4) {
            v2f Alo = *(const v2f*)(wklo + c0);            // b64, batch-invariant
            v2f Ahi = *(const v2f*)(wklo + 16 * CIN + c0); // b64, batch-invariant
#pragma unroll
            for (int bb = 0; bb < BATCH; ++bb) {
                // uniform SGPR base per batch + shared VGPR offset (GVS)
                v2f Bv = *(const v2f*)(in_t + bb * (NLEN * CIN) + boff + c0);
                acc[bb][0] = __builtin_amdgcn_wmma_f32_16x16x4_f32(
                    false, Alo, false, Bv, (short)0, acc[bb][0], false, false);
                acc[bb][1] = __builtin_amdgcn_wmma_f32_16x16x4_f32(
                    false, Ahi, false, Bv, (short)0, acc[bb][1], false, false);
            }
        }
    }

    // Epilogue: C/D layout -> VGPR r holds M = r + half*8, N = n0 + lcol.
    float* outCol = out + n0 + lcol;
#pragma unroll
    for (int bb = 0; bb < BATCH; ++bb) {
#pragma unroll
        for (int r = 0; r < 8; ++r) {
            const int o_lo = r + half * 8;
            const int o_hi = o_lo + 16;
            float vlo = acc[bb][0][r] + bias[o_lo];
            float vhi = acc[bb][1][r] + bias[o_hi];
            outCol[(bb * COUT + o_lo) * NLEN] = fmaxf(vlo, 0.0f);
            outCol[(bb * COUT + o_hi) * NLEN] = fmaxf(vhi, 0.0f);
        }
    }
}

extern "C" void kernel_launch(void* const* d_in, const int* in_sizes, int n_in,
                              void* d_out, int out_size, void* d_ws, size_t ws_size,
                              hipStream_t stream) {
    const float* inp   = (const float*)d_in[0]; // (4, 32, 81920) f32
    const float* wgt   = (const float*)d_in[1]; // (32, 32, 10, 1) f32
    const float* bias  = (const float*)d_in[2]; // (32,) f32
    const int*   table = (const int*)d_in[3];   // (10, 81920) i32
    // d_in[4] = stride (== 1 in setup); stride 1 handled implicitly.
    float* out = (float*)d_out;                 // (4, 32, 1, 81920) f32

    // Workspace layout: [ input_t : 4*81920*32 f32 = 41.94 MB | wt : 40 KB ]
    float* in_t = (float*)d_ws;
    float* wt   = in_t + (size_t)BATCH * NLEN * CIN;

    PHD_transpose_input<<<dim3(BATCH * (NLEN / TT_N)), dim3(256), 0, stream>>>(inp, in_t);
    PHD_transpose_weight<<<dim3((KW * COUT * CIN + 255) / 256), dim3(256), 0, stream>>>(wgt, wt);

    PHD_conv_gather_wmma_f32<<<dim3(NBLK), dim3(32 * WPB), 0, stream>>>(
        in_t, wt, bias, table, out);
}